// PointNet_64072322122107
// MI455X (gfx1250) — compile-verified
//
#include <hip/hip_runtime.h>

typedef float v2f __attribute__((ext_vector_type(2)));
typedef float v8f __attribute__((ext_vector_type(8)));

// Explicit global-address-space pointers: force global_load/global_store
// (not flat_*) in the GEMM hot loop, so only LOADcnt/STOREcnt are involved.
typedef const __attribute__((address_space(1))) float* gcf;
typedef __attribute__((address_space(1))) float*       gf;
typedef const __attribute__((address_space(1))) v2f*   gcv2f;

// ---------------------------------------------------------------------------
// WMMA fp32 tile GEMM:  Y[b,o,n] = bias[o] + sum_c A(b,o,c) * X[b,c,n]
//   A(b,o,c) = TRANSW ? W[b*wbStride + c*ldw + o] : W[b*wbStride + o*ldw + c]
// One wave computes a 16 x (16*NT) output strip with V_WMMA_F32_16X16X4_F32
// (exact fp32 accumulation, matches the fp32 reference).
// REQUIREMENT: O is a multiple of 16 (all call sites satisfy this), grid.y=O/16.
// Operand packing per CDNA5 ISA 7.12.2:
//   A 16x4 : lane = 16*(k>=2) + m, vgpr j = k&1  ->  a[j] = A(o0+ln, c0+2*hf+j)
//   B 4x16 : vgpr j holds row k = 2*hf + j       ->  b[j] = X(c0+2*hf+j, n0+ln)
//   C/D    : d[v] = Y(o0 + v + 8*hf, n0 + ln)
// Main K-loop is branch-free: C&~3 full steps (A via one 8B load when !TRANSW),
// then a single guarded tail step handles C%4 (the C=3 layers hit only the tail).
// ---------------------------------------------------------------------------
template <bool TRANSW, int NT>
__global__ __launch_bounds__(32) void k_gemm_wmma(
    const float* __restrict__ W, const float* __restrict__ bias,
    const float* __restrict__ X, float* __restrict__ Y,
    int O, int C, int N, int ldw, long wbStride)
{
  const int lane = threadIdx.x & 31;
  const int hf = lane >> 4;    // half-wave selector
  const int ln = lane & 15;
  const int n0 = blockIdx.x * (16 * NT);
  const int o0 = blockIdx.y * 16;
  const int b  = blockIdx.z;

  const gcf Wb   = (gcf)(W + (size_t)b * (size_t)wbStride);
  const gcf Xb   = (gcf)(X + (size_t)b * (size_t)C * (size_t)N);
  const gf  Yb   = (gf)(Y + (size_t)b * (size_t)O * (size_t)N);
  const gcf biag = (gcf)bias;

  const int orow = o0 + ln;                 // < O by grid construction

  // Walking A pointer: element (orow, c) lives at aBase[c * aStep].
  const size_t aStep = TRANSW ? (size_t)ldw : (size_t)1;
  gcf aCur =
      (TRANSW ? (Wb + orow) : (Wb + (size_t)orow * ldw)) + (size_t)(2 * hf) * aStep;

  int nIdx[NT];
  gcf xCur[NT];
#pragma unroll
  for (int t = 0; t < NT; ++t) {
    nIdx[t] = n0 + 16 * t + ln;
    int nc = nIdx[t] < N ? nIdx[t] : N - 1;          // clamp; cols >= N unstored
    xCur[t] = Xb + nc + (size_t)(2 * hf) * N;
  }

  v8f acc[NT];
#pragma unroll
  for (int t = 0; t < NT; ++t)
#pragma unroll
    for (int v = 0; v < 8; ++v) {
      int o = o0 + v + 8 * hf;
      acc[t][v] = (bias != nullptr) ? biag[o] : 0.0f;
    }

  const int Cmain = C & ~3;
  for (int c0 = 0; c0 < Cmain; c0 += 4) {
    v2f a;
    if (TRANSW) { a[0] = aCur[0]; a[1] = aCur[aStep]; }
    else        { a = *(gcv2f)aCur; }                // 8B aligned: ldw even here
    aCur += 4 * aStep;
#pragma unroll
    for (int t = 0; t < NT; ++t) {
      v2f bm;
      bm[0] = xCur[t][0];
      bm[1] = xCur[t][N];
      xCur[t] += (size_t)4 * N;
      acc[t] = __builtin_amdgcn_wmma_f32_16x16x4_f32(
          false, a, false, bm, (short)0, acc[t], false, false);
    }
  }

  if (Cmain < C) {                                   // tail: 1..3 channels
    v2f a, bm[NT];
#pragma unroll
    for (int j = 0; j < 2; ++j) {
      int cc  = Cmain + 2 * hf + j;
      int ccc = cc < C ? cc : C - 1;
      a[j] = TRANSW ? Wb[(size_t)ccc * ldw + orow]
                    : Wb[(size_t)orow * ldw + ccc];
#pragma unroll
      for (int t = 0; t < NT; ++t) {
        int nc = nIdx[t] < N ? nIdx[t] : N - 1;
        float bv = Xb[(size_t)ccc * N + nc];
        bm[t][j] = cc < C ? bv : 0.0f;               // zero B row kills k>=C terms
      }
    }
#pragma unroll
    for (int t = 0; t < NT; ++t)
      acc[t] = __builtin_amdgcn_wmma_f32_16x16x4_f32(
          false, a, false, bm[t], (short)0, acc[t], false, false);
  }

#pragma unroll
  for (int t = 0; t < NT; ++t) {
    if (nIdx[t] < N) {
#pragma unroll
      for (int v = 0; v < 8; ++v) {
        int o = o0 + v + 8 * hf;
        Yb[(size_t)o * N + nIdx[t]] = acc[t][v];
      }
    }
  }
}

// ---------------------------------------------------------------------------
// Training-mode BatchNorm (+ReLU) over (B,N) per channel, fused two-pass.
// One block per channel.
// ---------------------------------------------------------------------------
__global__ void k_bn_relu(float* __restrict__ Y, const float* __restrict__ gam,
                          const float* __restrict__ bet, int B, int C, int N)
{
  const int c = blockIdx.x;
  const int t = threadIdx.x, T = blockDim.x;
  __shared__ float s1[256], s2[256];
  float s = 0.f, q = 0.f;
  for (int b = 0; b < B; ++b) {
    const float* p = Y + ((size_t)b * C + c) * N;
    for (int n = t; n < N; n += T) { float v = p[n]; s += v; q += v * v; }
  }
  s1[t] = s; s2[t] = q;
  __syncthreads();
  for (int st = T >> 1; st > 0; st >>= 1) {
    if (t < st) { s1[t] += s1[t + st]; s2[t] += s2[t + st]; }
    __syncthreads();
  }
  const float inv = 1.0f / ((float)B * (float)N);
  const float m   = s1[0] * inv;
  const float var = s2[0] * inv - m * m;          // biased var (jnp.var)
  const float rs  = rsqrtf(var + 1e-5f);
  const float ga = gam[c], be = bet[c];
  for (int b = 0; b < B; ++b) {
    float* p = Y + ((size_t)b * C + c) * N;
    for (int n = t; n < N; n += T) {
      float v = (p[n] - m) * rs * ga + be;
      p[n] = fmaxf(v, 0.0f);
    }
  }
}

// BN (+ReLU) over the batch dim for [B,C] vectors; one thread per channel.
__global__ void k_bn_relu_vec(float* __restrict__ x, const float* __restrict__ gam,
                              const float* __restrict__ bet, int B, int C)
{
  int c = blockIdx.x * blockDim.x + threadIdx.x;
  if (c >= C) return;
  float s = 0.f, q = 0.f;
  for (int b = 0; b < B; ++b) { float v = x[(size_t)b * C + c]; s += v; q += v * v; }
  float m = s / B, var = q / B - m * m;
  float rs = rsqrtf(var + 1e-5f);
  float ga = gam[c], be = bet[c];
  for (int b = 0; b < B; ++b) {
    float v = (x[(size_t)b * C + c] - m) * rs * ga + be;
    x[(size_t)b * C + c] = fmaxf(v, 0.0f);
  }
}

// Global max-pool over N; one block per (b,c) row.
__global__ void k_rowmax(const float* __restrict__ Y, float* __restrict__ out, int N)
{
  const int row = blockIdx.x;
  const float* p = Y + (size_t)row * N;
  __shared__ float sm[256];
  float m = -3.4e38f;
  for (int n = threadIdx.x; n < N; n += blockDim.x) m = fmaxf(m, p[n]);
  sm[threadIdx.x] = m;
  __syncthreads();
  for (int s = blockDim.x >> 1; s > 0; s >>= 1) {
    if (threadIdx.x < s) sm[threadIdx.x] = fmaxf(sm[threadIdx.x], sm[threadIdx.x + s]);
    __syncthreads();
  }
  if (threadIdx.x == 0) out[row] = sm[0];
}

// FC layer y[b,o] = bias[o] + W[o,:]·x[b,:]  (+ identity when eyeK>0).
__global__ void k_linear(const float* __restrict__ W, const float* __restrict__ bias,
                         const float* __restrict__ x, float* __restrict__ y,
                         int B, int O, int C, int eyeK)
{
  int i = blockIdx.x * blockDim.x + threadIdx.x;
  if (i >= B * O) return;
  int b = i / O, o = i % O;
  const float* wr = W + (size_t)o * C;
  const float* xr = x + (size_t)b * C;
  float acc = bias[o];
  for (int c = 0; c < C; ++c) acc += wr[c] * xr[c];
  if (eyeK > 0 && (o % (eyeK + 1)) == 0) acc += 1.0f;  // + eye(k) flattened
  y[(size_t)b * O + o] = acc;
}

// Apply the 3x3 STN: out[b,e,n] = sum_c pt[b,c,n] * tr[b, c*3+e]
__global__ void k_stn_apply(const float* __restrict__ pt, const float* __restrict__ tr,
                            float* __restrict__ out, int B, int N)
{
  int i = blockIdx.x * blockDim.x + threadIdx.x;
  if (i >= B * 3 * N) return;
  int n = i % N;
  int e = (i / N) % 3;
  int b = i / (3 * N);
  const float* p = pt + (size_t)b * 3 * N;
  const float* t = tr + (size_t)b * 9;
  out[i] = p[n] * t[e] + p[N + n] * t[3 + e] + p[2 * N + n] * t[6 + e];
}

// Farthest-point sampling (deterministic start index 0, matches reference scan).
// One block per batch; dist[] kept in workspace.
__global__ void k_fps(const float* __restrict__ pts, float* __restrict__ outp,
                      float* __restrict__ dist, int N, int M)
{
  const int b = blockIdx.x;
  const int t = threadIdx.x, T = blockDim.x;
  const float* px = pts + (size_t)b * 3 * N;
  float* db = dist + (size_t)b * N;
  __shared__ float sb[256];
  __shared__ int   si[256];
  for (int n = t; n < N; n += T) db[n] = 1e10f;
  __syncthreads();
  int far = 0;
  for (int i = 0; i < M; ++i) {
    if (t == 0) {                       // scan emits the incoming `far`
      outp[(size_t)b * 3 * M + i]         = px[far];
      outp[(size_t)b * 3 * M + M + i]     = px[N + far];
      outp[(size_t)b * 3 * M + 2 * M + i] = px[2 * N + far];
    }
    float cx = px[far], cy = px[N + far], cz = px[2 * N + far];
    float best = -1.0f; int bi = 0;
    for (int n = t; n < N; n += T) {
      float dx = px[n] - cx, dy = px[N + n] - cy, dz = px[2 * N + n] - cz;
      float d = dx * dx + dy * dy + dz * dz;
      d = fminf(d, db[n]);
      db[n] = d;
      if (d > best) { best = d; bi = n; }
    }
    sb[t] = best; si[t] = bi;
    __syncthreads();
    for (int s = T >> 1; s > 0; s >>= 1) {
      if (t < s && sb[t + s] > sb[t]) { sb[t] = sb[t + s]; si[t] = si[t + s]; }
      __syncthreads();
    }
    far = si[0];
    __syncthreads();
  }
}

// Bilinear 2x upsample, align_corners=True.
__global__ void k_up2x(const float* __restrict__ src, float* __restrict__ dst,
                       int H, int W, int total)
{
  int i = blockIdx.x * blockDim.x + threadIdx.x;
  if (i >= total) return;
  const int W2 = 2 * W, H2 = 2 * H;
  int x  = i % W2;
  int y  = (i / W2) % H2;
  int bc = i / (W2 * H2);
  float sy = (H > 1) ? (float)(H - 1) / (float)(H2 - 1) : 0.f;
  float sx = (W > 1) ? (float)(W - 1) / (float)(W2 - 1) : 0.f;
  float fy = y * sy; int y0 = (int)fy; int y1 = min(y0 + 1, H - 1); float wy = fy - y0;
  float fx = x * sx; int x0 = (int)fx; int x1 = min(x0 + 1, W - 1); float wx = fx - x0;
  const float* s = src + (size_t)bc * H * W;
  float v00 = s[y0 * W + x0], v01 = s[y0 * W + x1];
  float v10 = s[y1 * W + x0], v11 = s[y1 * W + x1];
  dst[i] = v00 * (1 - wy) * (1 - wx) + v01 * (1 - wy) * wx
         + v10 * wy * (1 - wx)       + v11 * wy * wx;
}

// ---------------------------------------------------------------------------
// Host-side orchestration.
// Input flattening (dict insertion order):
//   0: pt ; 1..22: stn3d ; 23..48: e1 ; 49..74: e2 ; 75..100: e3 ; 101..126: e4
// stn3d: c1w,c1b,c2w,c2b,c3w,c3b,f1w,f1b,f2w,f2b,f3w,f3b,bn1g,bn1b,...,bn5g,bn5b
// enc:   c1w,c1b,c2w,c2b,bn1g,bn1b,bn2g,bn2b, then fstn:
//        c1w,c1b,c2w,c2b,c3w,c3b,f1w,f1b,f2w,f2b,bn1g,bn1b,bn2g,bn2b,bn3g,bn3b,bn4g,bn4b
// ---------------------------------------------------------------------------
extern "C" void kernel_launch(void* const* d_in, const int* in_sizes, int n_in,
                              void* d_out, int out_size, void* d_ws, size_t ws_size,
                              hipStream_t stream)
{
  (void)in_sizes; (void)n_in; (void)out_size; (void)ws_size;
  auto P = [&](int i) -> const float* { return (const float*)d_in[i]; };
  const float* pt0 = P(0);
  const int B = 8, N3 = 6400, N2 = 1600, N1 = 400, N0 = 100;

  float* ws = (float*)d_ws;
  size_t off = 0;
  auto alloc = [&](size_t n) { float* p = ws + off; off += n; return p; };

  float* ptT    = alloc((size_t)B * 3 * N3);
  float* bufA   = alloc((size_t)B * 64 * N3);     // enc h1 / stn h1
  float* bufB   = alloc((size_t)B * 64 * N3);     // fstn h1, later feat-transformed x
  float* bufC   = alloc((size_t)B * 128 * N3);    // 128-ch intermediates
  float* bufD   = alloc((size_t)B * 1024 * N3);   // 512/1024-ch intermediates
  float* pooled = alloc((size_t)B * 1024);
  float* fcA    = alloc((size_t)B * 512);
  float* fcB    = alloc((size_t)B * 256);
  float* tr9    = alloc((size_t)B * 9);
  float* tmat   = alloc((size_t)B * 4096);
  float* dist   = alloc((size_t)B * N3);
  float* pts2   = alloc((size_t)B * 3 * N2);
  float* pts1   = alloc((size_t)B * 3 * N1);
  float* pts0   = alloc((size_t)B * 3 * N0);
  float* p3     = alloc((size_t)B * 64 * N3);
  float* p2     = alloc((size_t)B * 128 * N2);
  float* p1     = alloc((size_t)B * 256 * N1);
  float* p0     = alloc((size_t)B * 512 * N0);

  auto gemm = [&](const float* Wm, const float* bias, const float* X, float* Y,
                  int O, int C, int N, int ldw, long wbs, bool tw) {
    dim3 g((N + 63) / 64, O / 16, B);   // O is a multiple of 16 at every call site
    if (tw)
      k_gemm_wmma<true, 4><<<g, dim3(32), 0, stream>>>(Wm, bias, X, Y, O, C, N, ldw, wbs);
    else
      k_gemm_wmma<false, 4><<<g, dim3(32), 0, stream>>>(Wm, bias, X, Y, O, C, N, ldw, wbs);
  };
  auto bnrelu = [&](float* Y, const float* g, const float* b2, int C, int N) {
    k_bn_relu<<<dim3(C), dim3(256), 0, stream>>>(Y, g, b2, B, C, N);
  };
  auto rowmax = [&](const float* Y, float* o, int rows, int N) {
    k_rowmax<<<dim3(rows), dim3(256), 0, stream>>>(Y, o, N);
  };
  auto linear = [&](const float* Wm, const float* bias, const float* x, float* y,
                    int O, int C, int eyeK) {
    int tot = B * O;
    k_linear<<<dim3((tot + 255) / 256), dim3(256), 0, stream>>>(Wm, bias, x, y, B, O, C, eyeK);
  };
  auto bnvec = [&](float* x, const float* g, const float* b2, int C) {
    k_bn_relu_vec<<<dim3((C + 255) / 256), dim3(256), 0, stream>>>(x, g, b2, B, C);
  };
  auto fps = [&](const float* src, float* dst, int N, int M) {
    k_fps<<<dim3(B), dim3(256), 0, stream>>>(src, dst, dist, N, M);
  };

  // ---- Stage 0: STN3d on raw points -> 3x3 transform, apply to points ----
  gemm(P(1), P(2), pt0, bufA, 64, 3, N3, 3, 0, false);        bnrelu(bufA, P(13), P(14), 64, N3);
  gemm(P(3), P(4), bufA, bufC, 128, 64, N3, 64, 0, false);    bnrelu(bufC, P(15), P(16), 128, N3);
  gemm(P(5), P(6), bufC, bufD, 1024, 128, N3, 128, 0, false); bnrelu(bufD, P(17), P(18), 1024, N3);
  rowmax(bufD, pooled, B * 1024, N3);
  linear(P(7),  P(8),  pooled, fcA, 512, 1024, 0);            bnvec(fcA, P(19), P(20), 512);
  linear(P(9),  P(10), fcA,    fcB, 256, 512, 0);             bnvec(fcB, P(21), P(22), 256);
  linear(P(11), P(12), fcB,    tr9, 9, 256, /*eyeK=*/3);
  {
    int tot = B * 3 * N3;
    k_stn_apply<<<dim3((tot + 255) / 256), dim3(256), 0, stream>>>(pt0, tr9, ptT, B, N3);
  }

  // ---- Encoder (PointNetEncoder w/ feature transform) ----
  auto enc = [&](const float* ptsIn, int N, int outC, int pb, float* outBuf) {
    gemm(P(pb + 0), P(pb + 1), ptsIn, bufA, 64, 3, N, 3, 0, false);
    bnrelu(bufA, P(pb + 4), P(pb + 5), 64, N);
    // fstn(64)
    gemm(P(pb + 8),  P(pb + 9),  bufA, bufB, 64, 64, N, 64, 0, false);
    bnrelu(bufB, P(pb + 18), P(pb + 19), 64, N);
    gemm(P(pb + 10), P(pb + 11), bufB, bufC, 128, 64, N, 64, 0, false);
    bnrelu(bufC, P(pb + 20), P(pb + 21), 128, N);
    gemm(P(pb + 12), P(pb + 13), bufC, bufD, 512, 128, N, 128, 0, false);
    bnrelu(bufD, P(pb + 22), P(pb + 23), 512, N);
    rowmax(bufD, pooled, B * 512, N);
    linear(P(pb + 14), P(pb + 15), pooled, fcA, 256, 512, 0);
    bnvec(fcA, P(pb + 24), P(pb + 25), 256);
    linear(P(pb + 16), P(pb + 17), fcA, tmat, 4096, 256, /*eyeK=*/64);
    // feature transform: per-batch 64x64 GEMM, A = t^T  (WMMA, TRANSW path)
    gemm(tmat, nullptr, bufA, bufB, 64, 64, N, 64, /*wbs=*/4096, true);
    // c2
    gemm(P(pb + 2), P(pb + 3), bufB, outBuf, outC, 64, N, 64, 0, false);
    bnrelu(outBuf, P(pb + 6), P(pb + 7), outC, N);
  };

  enc(ptT,  N3, 64,  101, p3);          // e4
  fps(ptT,  pts2, N3, N2);
  enc(pts2, N2, 128, 75,  p2);          // e3
  fps(pts2, pts1, N2, N1);
  enc(pts1, N1, 256, 49,  p1);          // e2
  fps(pts1, pts0, N1, N0);
  enc(pts0, N0, 512, 23,  p0);          // e1

  // ---- Bilinear 2x upsample into concatenated d_out ----
  float* outF = (float*)d_out;
  auto up = [&](const float* src, float* dst, int C, int H, int W) {
    int tot = B * C * 4 * H * W;
    k_up2x<<<dim3((tot + 255) / 256), dim3(256), 0, stream>>>(src, dst, H, W, tot);
  };
  up(p0, outF + 0,        512, 10, 10);   // -> 8x512x20x20   (1,638,400)
  up(p1, outF + 1638400,  256, 20, 20);   // -> 8x256x40x40   (3,276,800)
  up(p2, outF + 4915200,  128, 40, 40);   // -> 8x128x80x80   (6,553,600)
  up(p3, outF + 11468800,  64, 80, 80);   // -> 8x64x160x160  (13,107,200)
}